// SparseLinear_4415226380843
// MI455X (gfx1250) — compile-verified
//
#include <hip/hip_runtime.h>

// ---------------- problem constants (match reference) ----------------
#define NUM_INPUTS  8192
#define NUM_OUTPUTS 8192
#define BATCH       512
#define NUM_TILES   (NUM_OUTPUTS / 16)   // 512 tiles of 16 output columns

typedef float v2f __attribute__((ext_vector_type(2)));
typedef float v8f __attribute__((ext_vector_type(8)));

// ---------------- K0: zero scratch (counts/cursor/offsets/poff/bucket) -------
__global__ __launch_bounds__(256) void sl_zero_kernel(int* __restrict__ p, int n) {
    int i = blockIdx.x * 256 + threadIdx.x;
    if (i < n) p[i] = 0;
}

// ---------------- K1: transpose x[512][8192] -> xT[8192][512] ----------------
__global__ __launch_bounds__(256) void sl_transpose_kernel(const float* __restrict__ x,
                                                           float* __restrict__ xT) {
    __shared__ float tile[32][33];
    const int rBase = blockIdx.x * 32;   // input-feature dim
    const int bBase = blockIdx.y * 32;   // batch dim
    const int tx = threadIdx.x;          // 0..31
    const int ty = threadIdx.y;          // 0..7
#pragma unroll
    for (int i = 0; i < 32; i += 8)
        tile[ty + i][tx] = x[(size_t)(bBase + ty + i) * NUM_INPUTS + rBase + tx];
    __syncthreads();
#pragma unroll
    for (int i = 0; i < 32; i += 8)
        xT[(size_t)(rBase + ty + i) * BATCH + bBase + tx] = tile[tx][ty + i];
}

// ---------------- K2: histogram of edge columns ----------------
__global__ __launch_bounds__(256) void sl_hist_kernel(const int* __restrict__ cols,
                                                      int* __restrict__ counts, int nnz) {
    int e = blockIdx.x * 256 + threadIdx.x;
    if (e < nnz) atomicAdd(&counts[cols[e]], 1);
}

// ---------------- K3: exclusive scan of counts -> offsets[NUM_OUTPUTS+1] -----
__global__ __launch_bounds__(1024) void sl_scan_kernel(const int* __restrict__ counts,
                                                       int* __restrict__ offsets) {
    __shared__ int s[1024];
    const int tid  = threadIdx.x;
    const int base = tid * 8;            // 1024 threads * 8 = 8192
    int c[8];
    int sum = 0;
#pragma unroll
    for (int i = 0; i < 8; ++i) { c[i] = counts[base + i]; sum += c[i]; }
    s[tid] = sum;
    __syncthreads();
    for (int off = 1; off < 1024; off <<= 1) {
        int t = s[tid];
        if (tid >= off) t += s[tid - off];
        __syncthreads();
        s[tid] = t;
        __syncthreads();
    }
    int run = (tid > 0) ? s[tid - 1] : 0;
#pragma unroll
    for (int i = 0; i < 8; ++i) { offsets[base + i] = run; run += c[i]; }
    if (tid == 1023) offsets[NUM_OUTPUTS] = run;
}

// ---------------- K3b: per-tile padded scan -> poff[NUM_TILES+1] -------------
// Each 16-col tile gets a bucket region rounded up to a multiple of 4 edges so
// the hot loop needs no bounds checks (pad slots are zeroed => w = 0).
__global__ __launch_bounds__(NUM_TILES) void sl_tile_scan_kernel(const int* __restrict__ offsets,
                                                                 int* __restrict__ poff) {
    __shared__ int s[NUM_TILES];
    const int t   = threadIdx.x;
    const int cnt = offsets[(t << 4) + 16] - offsets[t << 4];
    const int pc  = (cnt + 3) & ~3;
    s[t] = pc;
    __syncthreads();
    for (int off = 1; off < NUM_TILES; off <<= 1) {
        int v = s[t];
        if (t >= off) v += s[t - off];
        __syncthreads();
        s[t] = v;
        __syncthreads();
    }
    poff[t] = s[t] - pc;                       // exclusive padded start
    if (t == NUM_TILES - 1) poff[NUM_TILES] = s[t];
}

// ---------------- K4: scatter edges into packed padded buckets ---------------
// bucket[p] = { row | (localcol << 16), float_bits(w) }
__global__ __launch_bounds__(256) void sl_scatter_kernel(const int* __restrict__ rows,
                                                         const int* __restrict__ cols,
                                                         const float* __restrict__ w,
                                                         const int* __restrict__ offsets,
                                                         const int* __restrict__ poff,
                                                         int* __restrict__ cursor,
                                                         int2* __restrict__ bucket, int nnz) {
    int e = blockIdx.x * 256 + threadIdx.x;
    if (e < nnz) {
        int c = cols[e];
        int t = c >> 4;
        int p = poff[t] + (offsets[c] - offsets[t << 4]) + atomicAdd(&cursor[c], 1);
        int2 m;
        m.x = rows[e] | ((c & 15) << 16);
        m.y = __float_as_int(w[e]);
        bucket[p] = m;
    }
}

// ---------------- K5: gather + WMMA accumulate ----------------
// One workgroup per 16-output-column tile; 8 waves x 4 M-tiles = 512 batch rows.
// Chunk of 4 edges: B(4x16) one-hot-scaled weights, A(16x4) gathered xT rows,
// V_WMMA_F32_16X16X4_F32 accumulation. No divergence in the hot loop; the next
// chunk's 16B metadata load is software-pipelined over the gathers + WMMAs.
__global__ __launch_bounds__(256) void sl_spmm_wmma_kernel(const float* __restrict__ xT,
                                                           const float* __restrict__ bias,
                                                           const int* __restrict__ poff,
                                                           const int2* __restrict__ bucket,
                                                           float* __restrict__ out) {
    const int tile = blockIdx.x;
    const int c0   = tile * 16;
    const int lane = threadIdx.x & 31;
    const int wave = threadIdx.x >> 5;       // 0..7
    const int n    = lane & 15;              // N index within tile
    const int half = lane >> 4;              // 0: K pair (0,1)  1: K pair (2,3)

    const int pstart = poff[tile];
    const int pend   = poff[tile + 1];       // pend - pstart is a multiple of 4

    // Fold bias into the accumulators.
    const float bv = bias[c0 + n];
    v8f acc[4];
#pragma unroll
    for (int j = 0; j < 4; ++j)
#pragma unroll
        for (int i = 0; i < 8; ++i) acc[j][i] = bv;

    // This lane's two K slots live at (e + half*2, e + half*2 + 1): one b128 load.
    const int2* __restrict__ mp = bucket + half * 2;
    int4 m = *(const int4*)(mp + pstart);    // pre-load first chunk's metadata

    for (int e = pstart; e < pend; e += 4) {
        // Software pipeline: issue next chunk's metadata load immediately
        // (tail over-read lands in extra allocated pad; value is discarded).
        int4 mn = *(const int4*)(mp + e + 4);
        __builtin_prefetch((const void*)(mp + e + 16), 0, 0);

        // Decode current chunk (rows 0..8191 in low 16 bits, lcol in high bits).
        const int   rA = m.x & 0xFFFF;
        const int   cA = m.x >> 16;
        const float wA = __int_as_float(m.y);
        const int   rB = m.z & 0xFFFF;
        const int   cB = m.z >> 16;
        const float wB = __int_as_float(m.w);

        // B matrix (4x16): B[k][n] = w_k iff edge k targets local column n.
        v2f bmat;
        bmat.x = (cA == n) ? wA : 0.0f;
        bmat.y = (cB == n) ? wB : 0.0f;

        const float* __restrict__ pA = xT + (size_t)rA * BATCH;
        const float* __restrict__ pB = xT + (size_t)rB * BATCH;

#pragma unroll
        for (int j = 0; j < 4; ++j) {
            const int b0 = (wave * 4 + j) * 16;  // batch base of this M-tile
            v2f amat;                            // A[16x4]: lane m = n, K = half*2(+1)
            amat.x = pA[b0 + n];                 // coalesced 64B per half-wave
            amat.y = pB[b0 + n];
            acc[j] = __builtin_amdgcn_wmma_f32_16x16x4_f32(
                false, amat, false, bmat, (short)0, acc[j], false, false);
        }
        m = mn;
    }

    // D layout: VGPR i -> M = i + half*8, lane -> N = n. Store y[b][c].
#pragma unroll
    for (int j = 0; j < 4; ++j) {
        const int b0 = (wave * 4 + j) * 16 + half * 8;
        float* __restrict__ op = out + (size_t)b0 * NUM_OUTPUTS + c0 + n;
#pragma unroll
        for (int i = 0; i < 8; ++i)
            op[(size_t)i * NUM_OUTPUTS] = acc[j][i];
    }
}

// ---------------- launch ----------------
extern "C" void kernel_launch(void* const* d_in, const int* in_sizes, int n_in,
                              void* d_out, int out_size, void* d_ws, size_t ws_size,
                              hipStream_t stream) {
    const float* x    = (const float*)d_in[0];   // [512, 8192]
    const float* w    = (const float*)d_in[1];   // [nnz]
    const float* bias = (const float*)d_in[2];   // [8192]
    const int*   rows = (const int*)d_in[3];     // [nnz]
    const int*   cols = (const int*)d_in[4];     // [nnz]
    const int    nnz  = in_sizes[1];

    // Padded bucket capacity: per-tile round-up (<=3 each) + pipeline tail pad.
    const int nbucket = nnz + NUM_TILES * 4 + 16;

    // Workspace layout.
    char* ws = (char*)d_ws;
    float* xT      = (float*)ws;  ws += (size_t)NUM_INPUTS * BATCH * sizeof(float);
    int*   counts  = (int*)ws;    ws += NUM_OUTPUTS * sizeof(int);
    int*   cursor  = (int*)ws;    ws += NUM_OUTPUTS * sizeof(int);
    int*   offsets = (int*)ws;    ws += (NUM_OUTPUTS + 64) * sizeof(int);
    int*   poff    = (int*)ws;    ws += (NUM_TILES + 64) * sizeof(int);
    int2*  bucket  = (int2*)ws;   // 8-byte aligned by construction

    float* out = (float*)d_out;

    // K0: zero everything from counts through the end of bucket (one range).
    const int zero_ints = (NUM_OUTPUTS * 2 + (NUM_OUTPUTS + 64) + (NUM_TILES + 64))
                          + 2 * nbucket;
    sl_zero_kernel<<<(zero_ints + 255) / 256, 256, 0, stream>>>(counts, zero_ints);

    // K1: transpose x -> xT (coalesced gather rows for the hot kernel)
    sl_transpose_kernel<<<dim3(NUM_INPUTS / 32, BATCH / 32), dim3(32, 8), 0, stream>>>(x, xT);

    // K2: per-column histogram
    sl_hist_kernel<<<(nnz + 255) / 256, 256, 0, stream>>>(cols, counts, nnz);

    // K3: column offsets
    sl_scan_kernel<<<1, 1024, 0, stream>>>(counts, offsets);

    // K3b: padded per-tile offsets
    sl_tile_scan_kernel<<<1, NUM_TILES, 0, stream>>>(offsets, poff);

    // K4: scatter packed edge metadata into padded buckets
    sl_scatter_kernel<<<(nnz + 255) / 256, 256, 0, stream>>>(rows, cols, w, offsets, poff,
                                                             cursor, bucket, nnz);

    // K5: gather + WMMA accumulate + bias -> d_out
    sl_spmm_wmma_kernel<<<NUM_TILES, 256, 0, stream>>>(xT, bias, poff, bucket, out);
}